// ConvModel_25202868093530
// MI455X (gfx1250) — compile-verified
//
#include <hip/hip_runtime.h>

typedef __attribute__((ext_vector_type(2))) float v2f;
typedef __attribute__((ext_vector_type(4))) float v4f;
typedef __attribute__((ext_vector_type(8))) float v8f;

#define N_    32
#define CIN   3
#define H_    224
#define W_    224
#define COUT  64
#define KREAL 27          // CIN * 3 * 3
#define HW    (H_ * W_)   // 50176
#define NPIX  (N_ * HW)   // 1,605,632

#define PIX_TILES  (NPIX / 16)   // 100,352 (one wave per pixel tile, all 64 c_out)
#define WAVES_PER_BLOCK 8
#define BLOCKS     (PIX_TILES / WAVES_PER_BLOCK)  // 12,544

static_assert(HW % 16 == 0, "pixel tiles must not cross image boundary");
static_assert(W_ % 16 == 0, "pixel tiles must not cross row boundary");
static_assert(PIX_TILES % WAVES_PER_BLOCK == 0, "grid must divide evenly");

__device__ __forceinline__ int imax(int a, int b) { return a > b ? a : b; }
__device__ __forceinline__ int imin(int a, int b) { return a < b ? a : b; }

// Implicit-GEMM conv via V_WMMA_F32_16X16X4_F32 (exact fp32, matches reference).
// One wave = 64(c_out) x 16(pixel) outputs: 4 accumulators, 7 K-steps x 4 WMMAs,
// one shared B fragment per step. K = 27 padded to 28; the k=27 column is zeroed
// on the A side only, so B needs no validity logic.
// Interior tiles (~85%, wave-uniform test) use pure constant-offset loads.
__global__ __launch_bounds__(256)
void conv3x3_wmma_f32(const float* __restrict__ x,
                      const float* __restrict__ w,
                      const float* __restrict__ bias,
                      float* __restrict__ out) {
    const int lane    = threadIdx.x & 31;
    const int wib     = threadIdx.x >> 5;
    const int pixTile = blockIdx.x * WAVES_PER_BLOCK + wib;   // wave-uniform

    const int half = lane >> 4;   // 0: K-pair {0,1}; 1: K-pair {2,3} per 4-chunk
    const int l16  = lane & 15;   // B: N index; A: M index

    // Tile coordinates (wave-uniform): one row segment of one image.
    const int pBase = pixTile * 16;
    const int n     = pBase / HW;
    const int spB   = pBase - n * HW;
    const int ohT   = spB / W_;          // row (same for all 16 pixels)
    const int owB   = spB - ohT * W_;    // column base (multiple of 16)

    const int sp    = spB + l16;               // per-lane spatial index
    const int xbase = n * (CIN * HW) + sp;     // per-lane base index into x

    // All 9 taps in-bounds for every pixel of the tile?
    const bool interior = (ohT >= 1) & (ohT <= H_ - 2) &
                          (owB >= 16) & (owB <= W_ - 32);

    // ---- B fragments for all 7 K-steps (shared across the 4 c_out tiles) ----
    v2f bf[7];

    if (interior) {
#pragma unroll
        for (int t = 0; t < 7; ++t) {
#pragma unroll
            for (int e = 0; e < 2; ++e) {
                const int kA = 4 * t + e;                  // half=0 k (const)
                const int kB = imin(kA + 2, KREAL - 1);    // half=1 k (k=27 dead)
                const int dA = (kA / 9) * HW + ((kA % 9) / 3 - 1) * W_ + ((kA % 9) % 3 - 1);
                const int dB = (kB / 9) * HW + ((kB % 9) / 3 - 1) * W_ + ((kB % 9) % 3 - 1);
                const float val = x[xbase + (half ? dB : dA)];
                if (e == 0) bf[t].x = val; else bf[t].y = val;
            }
        }
    } else {
        // Clamped spatial offsets + float 0/1 masks (no bool arrays).
        int   spOff[9];
        float mf[9];
#pragma unroll
        for (int kh = 0; kh < 3; ++kh) {
#pragma unroll
            for (int kw = 0; kw < 3; ++kw) {
                const int ih = ohT + kh - 1;           // wave-uniform
                const int iw = owB + l16 + kw - 1;     // per-lane
                const int i  = kh * 3 + kw;
                mf[i] = (((unsigned)ih < (unsigned)H_) &
                         ((unsigned)iw < (unsigned)W_)) ? 1.0f : 0.0f;
                spOff[i] = imin(imax(ih, 0), H_ - 1) * W_ +
                           imin(imax(iw, 0), W_ - 1);
            }
        }
        const int nbase = n * (CIN * HW);
#pragma unroll
        for (int t = 0; t < 7; ++t) {
#pragma unroll
            for (int e = 0; e < 2; ++e) {
                const int kA  = 4 * t + e;
                const int kB  = imin(kA + 2, KREAL - 1);
                const int ciA = kA / 9, rA = kA % 9;
                const int ciB = kB / 9, rB = kB % 9;
                const int off = (half ? (ciB * HW) : (ciA * HW)) +
                                (half ? spOff[rB] : spOff[rA]);
                float val = x[nbase + off];            // always-valid address
                val *= half ? mf[rB] : mf[rA];         // zero OOB taps
                if (e == 0) bf[t].x = val; else bf[t].y = val;
            }
        }
    }

    // ---- 7 K-steps x 4 c_out tiles; independent acc chains back-to-back ----
    v8f acc[4] = {v8f{}, v8f{}, v8f{}, v8f{}};

#pragma unroll
    for (int t = 0; t < 7; ++t) {
#pragma unroll
        for (int c = 0; c < 4; ++c) {
            const float* __restrict__ wrh =
                w + (c * 16 + l16) * KREAL + 2 * half;   // this lane's K-pair base
            v2f a;
            a.x = wrh[4 * t];                            // k = 4t+2h   (<= 26)
            if (t == 6) {
                // k of a.y = 27 for half=1: load k=26 instead, then zero it.
                float ay = wrh[25 - half];               // k = 25 (h0) / 26 (h1)
                a.y = half ? 0.0f : ay;
            } else {
                a.y = wrh[4 * t + 1];                    // k = 4t+2h+1 (<= 25)
            }
            acc[c] = __builtin_amdgcn_wmma_f32_16x16x4_f32(
                false, a, false, bf[t], (short)0, acc[c], false, false);
        }
    }

    // ---- Store: VGPR v of acc[c] holds (co = c*16 + 8*half + v, pixel l16) ----
    const int outBase = n * (COUT * HW) + sp;
#pragma unroll
    for (int c = 0; c < 4; ++c) {
        const int co0 = c * 16 + 8 * half;
        const v4f* bp = (const v4f*)(bias + co0);        // 16B-aligned
        const v4f b0 = bp[0], b1 = bp[1];
        float* __restrict__ op = out + outBase + co0 * HW;
#pragma unroll
        for (int v = 0; v < 8; ++v) {
            const float bv = (v < 4) ? b0[v] : b1[v - 4];
            op[v * HW] = acc[c][v] + bv;                 // imm offsets < 8MB
        }
    }
}

extern "C" void kernel_launch(void* const* d_in, const int* in_sizes, int n_in,
                              void* d_out, int out_size, void* d_ws, size_t ws_size,
                              hipStream_t stream) {
    const float* x    = (const float*)d_in[0];
    const float* w    = (const float*)d_in[1];
    const float* bias = (const float*)d_in[2];
    float* out        = (float*)d_out;

    conv3x3_wmma_f32<<<dim3(BLOCKS), dim3(256), 0, stream>>>(x, w, bias, out);
}